// BoundaryLoss_65008624992505
// MI455X (gfx1250) — compile-verified
//
#include <hip/hip_runtime.h>
#include <stdint.h>

#define BATCH 32
#define IMH   1024
#define IMW   1024
#define TW    128   // tile width  (interior)
#define TH    32    // tile height (interior)
#define LW    132   // loaded window width  (TW + 4)
#define LH    36    // loaded window height (TH + 4)
#define ES    136   // erode/dilate LDS row stride (bytes)
#define EROWS 34    // TH + 2
#define ECOLS 130   // TW + 2
#define NTHREADS 256
#define NTILES_X (IMW / TW)                      // 8
#define NTILES_Y (IMH / TH)                      // 32
#define NBLOCKS  (NTILES_X * NTILES_Y * BATCH)   // 8192
#define NELEM    ((float)BATCH * IMH * IMW)

typedef uint32_t u32x4 __attribute__((ext_vector_type(4)));
typedef uint32_t u32x8 __attribute__((ext_vector_type(8)));

// Low 32 bits of a generic pointer into __shared__ == LDS byte address
// (ISA 10.2: the LDS aperture address is the low 32 address bits).
__device__ __forceinline__ uint32_t lds_addr32(const void* p) {
    return (uint32_t)(uintptr_t)p;
}

// mask bit with image-boundary zero padding (unsigned compare handles <0)
__device__ __forceinline__ unsigned mbit(const float* sRaw, int gx, int gy,
                                         int xs, int ys) {
    if ((unsigned)gx >= IMW || (unsigned)gy >= IMH) return 0u;
    return sRaw[(gy - ys) * LW + (gx - xs)] > 0.5f ? 1u : 0u;
}

__global__ void __launch_bounds__(NTHREADS)
boundary_bce_partial(const float* __restrict__ inp,
                     const float* __restrict__ tgt,
                     float* __restrict__ partial)
{
    __shared__ __align__(16) float         sRaw[LH * LW];   // TDM-filled targets window
    __shared__ __align__(16) unsigned char sED[EROWS * ES]; // bit0=erode1, bit1=dilate1
    __shared__ float sRed[8];

    const int tid = threadIdx.x;
    const int wg  = blockIdx.x;
    const int tx  = wg & (NTILES_X - 1);
    const int ty  = (wg >> 3) & (NTILES_Y - 1);
    const int b   = wg >> 8;
    const int x0  = tx * TW;
    const int y0  = ty * TH;
    const int imgBase = b * (IMH * IMW);   // fits in int

    // Clamped in-bounds LW x LH window (slides inward at image borders)
    const int xs = min(max(x0 - 2, 0), IMW - LW);
    const int ys = min(max(y0 - 2, 0), IMH - LH);
    const int ix0 = x0 - xs;               // window col of tile-interior col 0 (in {0,2,4})
    const int iy0 = y0 - ys;               // window row of tile-interior row 0
    const bool edgeB = (tx == 0) | (tx == NTILES_X - 1) | (ty == 0) | (ty == NTILES_Y - 1);

    // ---- Phase 1: one Tensor-Data-Mover DMA fills the targets window in LDS.
    // Descriptor: 2D tile == tensor (132x36, no OOB path), row pitch 1024 elems.
    if (tid == 0) {
        const uint64_t ga = (uint64_t)(uintptr_t)tgt +
                            ((uint64_t)(imgBase + ys * IMW + xs) << 2);
        u32x4 g0;
        g0.x = 1u;                                   // count=1, user descriptor
        g0.y = lds_addr32(&sRaw[0]);                 // lds_addr [63:32]
        g0.z = (uint32_t)ga;                         // global_addr [95:64]
        g0.w = (uint32_t)((ga >> 32) & 0x01FFFFFFu)  // global_addr [120:96]
             | (2u << 30);                           // type=2 ("image")
        u32x8 g1;
        g1.s0 = (2u << 16);                          // data_size = 4 bytes
        g1.s1 = ((uint32_t)LW << 16);                // tensor_dim0[15:0] @ bits 63:48
        g1.s2 = ((uint32_t)LH << 16);                // dim0 hi=0 | tensor_dim1[15:0]
        g1.s3 = ((uint32_t)LW << 16);                // dim1 hi=0 | tile_dim0
        g1.s4 = (uint32_t)LH;                        // tile_dim1 | tile_dim2=0
        g1.s5 = (uint32_t)IMW;                       // tensor_dim0_stride[31:0]
        g1.s6 = 0u;                                  // stride0 hi | stride1 lo
        g1.s7 = 0u;                                  // stride1 hi
        const u32x4 gz = {0u, 0u, 0u, 0u};           // groups 2/3 unused (2D)
        asm volatile("tensor_load_to_lds %0, %1, %2, %3"
                     :: "s"(g0), "s"(g1), "s"(gz), "s"(gz) : "memory");
    }

    // ---- Phase 2 (overlapped with TDM): stream this thread's inputs (b128).
    const int cx  = (tid & 31) * 4;        // lane-consecutive -> coalesced 512B rows
    const int ry0 = (tid >> 5) * 4;
    float4 iv[4];
    #pragma unroll
    for (int j = 0; j < 4; ++j)
        iv[j] = *(const float4*)&inp[imgBase + (y0 + ry0 + j) * IMW + x0 + cx];

#if __has_builtin(__builtin_amdgcn_s_wait_tensorcnt)
    __builtin_amdgcn_s_wait_tensorcnt(0);
#else
    asm volatile("s_wait_tensorcnt 0" ::: "memory");
#endif
    __syncthreads();

    // ---- Phase 3: pass-1 cross erode/dilate of mask, packed 2 bits/pixel.
    if (!edgeB) {
        for (int t = tid; t < EROWS * ECOLS; t += NTHREADS) {
            const int py = t / ECOLS;
            const int px = t % ECOLS;
            const float* rc = &sRaw[(py + 1) * LW + (px + 1)]; // iy0==ix0==2 here
            const unsigned c = rc[0]   > 0.5f;
            const unsigned u = rc[-LW] > 0.5f;
            const unsigned d = rc[LW]  > 0.5f;
            const unsigned l = rc[-1]  > 0.5f;
            const unsigned r = rc[1]   > 0.5f;
            sED[py * ES + px] =
                (unsigned char)((c & u & d & l & r) | ((c | u | d | l | r) << 1));
        }
    } else {
        for (int t = tid; t < EROWS * ECOLS; t += NTHREADS) {
            const int py = t / ECOLS;
            const int px = t % ECOLS;
            const int gy = y0 - 1 + py;    // raw row py+1 -> global
            const int gx = x0 - 1 + px;
            const unsigned c = mbit(sRaw, gx,     gy,     xs, ys);
            const unsigned u = mbit(sRaw, gx,     gy - 1, xs, ys);
            const unsigned d = mbit(sRaw, gx,     gy + 1, xs, ys);
            const unsigned l = mbit(sRaw, gx - 1, gy,     xs, ys);
            const unsigned r = mbit(sRaw, gx + 1, gy,     xs, ys);
            sED[py * ES + px] =
                (unsigned char)((c & u & d & l & r) | ((c | u | d | l | r) << 1));
        }
    }
    __syncthreads();

    // ---- Phase 4: pass-2 morphology + weighted BCE, accumulate.
    float acc = 0.0f;
    #pragma unroll
    for (int j = 0; j < 4; ++j) {
        const int y = ry0 + j;
        const float ivv[4] = {iv[j].x, iv[j].y, iv[j].z, iv[j].w};
        #pragma unroll
        for (int q = 0; q < 4; ++q) {
            const int x = cx + q;
            const unsigned c  = sED[(y + 1) * ES + (x + 1)];
            const unsigned u  = sED[ y      * ES + (x + 1)];
            const unsigned dd = sED[(y + 2) * ES + (x + 1)];
            const unsigned l  = sED[(y + 1) * ES +  x     ];
            const unsigned r  = sED[(y + 1) * ES + (x + 2)];
            const unsigned e2 = c & u & dd & l & r & 1u;
            const unsigned d2 = ((c | u | dd | l | r) >> 1) & 1u;
            const float wgt = 1.0f + 5.0f * (float)(d2 & (e2 ^ 1u));
            const float tv  = sRaw[(iy0 + y) * LW + (ix0 + x)]; // always in-window
            const float xv  = ivv[q];
            // logaddexp(0,x) = max(x,0) + log1p(exp(-|x|))
            const float bce = fmaxf(xv, 0.0f) + log1pf(expf(-fabsf(xv))) - tv * xv;
            acc += bce * wgt;
        }
    }

    // ---- Phase 5: deterministic reduction (wave32 xor-tree, then LDS).
    #pragma unroll
    for (int off = 16; off > 0; off >>= 1)
        acc += __shfl_xor(acc, off, 32);
    if ((tid & 31) == 0) sRed[tid >> 5] = acc;
    __syncthreads();
    if (tid == 0) {
        float s = 0.0f;
        #pragma unroll
        for (int i = 0; i < 8; ++i) s += sRed[i];
        partial[wg] = s;
    }
}

__global__ void __launch_bounds__(NTHREADS)
reduce_partials(const float* __restrict__ partial, float* __restrict__ out, int n)
{
    __shared__ float sm[NTHREADS];
    const int tid = threadIdx.x;
    float s = 0.0f;
    for (int i = tid; i < n; i += NTHREADS) s += partial[i];
    sm[tid] = s;
    __syncthreads();
    for (int k = NTHREADS / 2; k > 0; k >>= 1) {
        if (tid < k) sm[tid] += sm[tid + k];
        __syncthreads();
    }
    if (tid == 0) out[0] = sm[0] * (1.0f / NELEM);
}

extern "C" void kernel_launch(void* const* d_in, const int* in_sizes, int n_in,
                              void* d_out, int out_size, void* d_ws, size_t ws_size,
                              hipStream_t stream)
{
    const float* inp = (const float*)d_in[0];
    const float* tgt = (const float*)d_in[1];
    float* partial   = (float*)d_ws;         // NBLOCKS floats = 32 KB scratch

    boundary_bce_partial<<<NBLOCKS, NTHREADS, 0, stream>>>(inp, tgt, partial);
    reduce_partials<<<1, NTHREADS, 0, stream>>>(partial, (float*)d_out, NBLOCKS);
}